// CausalSelfAttention_51977694216452
// MI455X (gfx1250) — compile-verified
//
#include <hip/hip_runtime.h>
#include <hip/hip_bf16.h>

// ---------------------------------------------------------------------------
// CDNA5 (gfx1250) f32 WMMA attention decode:
//   q/k/v = x @ w{q,k,v}  (WMMA f32 16x16x4 GEMM, compile-time N)
//   RoPE(q,k) (q pre-scaled by 1/sqrt(dh))
//   flash attention over sink(4) + sliding window(2048) keys
//     (double-buffered async global->LDS K/V staging, WMMA QK^T and PV)
//   out = y @ w_proj
// ---------------------------------------------------------------------------

typedef __attribute__((ext_vector_type(2))) float v2f;
typedef __attribute__((ext_vector_type(8))) float v8f;
typedef int i32x4 __attribute__((vector_size(16)));

#define B_SZ     16
#define T_SZ     4
#define C_SZ     2048
#define NH       16
#define NKV      4
#define DH       128
#define GROUP    4          // NH / NKV
#define S_CACHE  4096
#define SINK     4
#define WINDOW   2048
#define SEFF     (SINK + WINDOW)      // 2052 effective key positions
#define NTILES   ((SEFF + 15) / 16)   // 129
#define NEG_BIG  (-3.0e38f)
#define LROW     132                  // padded LDS row stride (floats)

#if defined(__AMDGCN__) &&                                                    \
    __has_builtin(__builtin_amdgcn_global_load_async_to_lds_b128) &&          \
    __has_builtin(__builtin_amdgcn_s_wait_asynccnt)
#define USE_ASYNC 1
#else
#define USE_ASYNC 0
#endif

#if USE_ASYNC
// Async copy 16 bytes global -> LDS (tracked by ASYNCcnt).
__device__ __forceinline__ void async_cp16(const float* g, float* l) {
  __attribute__((address_space(1))) i32x4* gp =
      (__attribute__((address_space(1))) i32x4*)(unsigned long long)
          (uintptr_t)g;
  __attribute__((address_space(3))) i32x4* lp =
      (__attribute__((address_space(3))) i32x4*)(unsigned int)(uintptr_t)l;
  __builtin_amdgcn_global_load_async_to_lds_b128(gp, lp, 0, 0);
}
#endif

// ---------------------------------------------------------------------------
// Generic GEMM: Out[64 x N] = X[64 x 2048] @ W[2048 x N], N compile-time.
// grid = (N/16, 64/16), block = 32 threads (one wave per 16x16 tile)
// ---------------------------------------------------------------------------
template <int N>
__global__ void wmma_gemm_f32(const float* __restrict__ X,
                              const float* __restrict__ W,
                              float* __restrict__ Out) {
  const int nb   = blockIdx.x;
  const int mb   = blockIdx.y;
  const int lane = threadIdx.x;
  const int l16  = lane & 15;
  const int half = lane >> 4;

  const float* xrow = X + (size_t)(mb * 16 + l16) * C_SZ + 2 * half;
  const float* wcol = W + (size_t)(2 * half) * N + nb * 16 + l16;

  v8f acc = {};
  for (int kb = 0; kb < C_SZ; kb += 16) {
    v2f a[4], b[4];
    #pragma unroll
    for (int u = 0; u < 4; ++u) {
      a[u] = *(const v2f*)(xrow + kb + 4 * u);
      const float* wp = wcol + (size_t)(kb + 4 * u) * N;
      b[u].x = wp[0];
      b[u].y = wp[N];
    }
    int pk = kb + 128;
    if (pk > C_SZ - 16) pk = C_SZ - 16;
    __builtin_prefetch(wcol + (size_t)pk * N, 0, 1);
    #pragma unroll
    for (int u = 0; u < 4; ++u)
      acc = __builtin_amdgcn_wmma_f32_16x16x4_f32(
          false, a[u], false, b[u], (short)0, acc, false, false);
  }

  #pragma unroll
  for (int j = 0; j < 8; ++j)
    Out[(size_t)(mb * 16 + j + 8 * half) * N + nb * 16 + l16] = acc[j];
}

// ---------------------------------------------------------------------------
// In-place RoPE over a (64 rows x heads*128) buffer, with optional output
// scaling (used to fold 1/sqrt(dh) into q). pos = start_pos + (row % T).
// ---------------------------------------------------------------------------
__global__ void rope_kernel(float* __restrict__ buf, int heads,
                            const int* __restrict__ start_pos, float scale) {
  int idx = blockIdx.x * blockDim.x + threadIdx.x;
  int total = 64 * heads * 64;
  if (idx >= total) return;
  int i = idx & 63;              // rotation pair index 0..63
  int h = (idx >> 6) % heads;
  int r = idx / (64 * heads);    // row 0..63 (b*T + t)
  int t = r & (T_SZ - 1);

  float pos = (float)(start_pos[0] + t);
  float ang = pos * __expf(-(float)i * (9.210340371976184f / 64.0f));
  float s, c;
  __sincosf(ang, &s, &c);

  float* p = buf + (size_t)r * heads * DH + h * DH + i;
  float x1 = p[0], x2 = p[64];
  p[0]  = (x1 * c - x2 * s) * scale;
  p[64] = (x1 * s + x2 * c) * scale;
}

// ---------------------------------------------------------------------------
// Flash attention: one wave per (batch, kv_head). 16 query rows
// (M = g*4 + t), 2052 keys in 16-wide tiles, d_head = 128.
// Q lives in LDS (no register spills); K/V tiles double-buffered via the
// async-to-LDS engine.
// ---------------------------------------------------------------------------
__global__ void attn_kernel(const float* __restrict__ q,
                            const float* __restrict__ knew,
                            const float* __restrict__ vnew,
                            const float* __restrict__ cache_k,
                            const float* __restrict__ cache_v,
                            float* __restrict__ y) {
  __shared__ float qtile[16 * LROW];
  __shared__ float ktile[2][16 * LROW];
  __shared__ float vtile[2][16 * LROW];
  __shared__ float ptile[16 * 17];

  const int wid  = blockIdx.x;
  const int b    = wid >> 2;
  const int kv   = wid & 3;
  const int lane = threadIdx.x;
  const int l16  = lane & 15;
  const int half = lane >> 4;
  const int ky   = lane >> 1;    // staging: row handled by this lane
  const int hh   = lane & 1;     // staging: which 64-dim half

  // ----- stage Q (16 rows x 128) into LDS, row M = g*4 + t -----
  {
    const int tq = ky & 3, gq = ky >> 2;
    const float* qp =
        q + ((size_t)(b * T_SZ + tq) * NH + (kv * GROUP + gq)) * DH + hh * 64;
    float* qd = &qtile[ky * LROW + hh * 64];
#if USE_ASYNC
    #pragma unroll
    for (int d4 = 0; d4 < 16; ++d4) async_cp16(qp + d4 * 4, qd + d4 * 4);
#else
    #pragma unroll
    for (int d4 = 0; d4 < 16; ++d4)
      *(float4*)(qd + d4 * 4) = *(const float4*)(qp + d4 * 4);
#endif
  }

  const float* kbase = cache_k + (size_t)(b * NKV + kv) * S_CACHE * DH;
  const float* vbase = cache_v + (size_t)(b * NKV + kv) * S_CACHE * DH;

  // ----- K/V tile staging: 2 lanes per key, 64 dims each. Out-of-range
  // keys use clamped addresses; their columns are masked below.
  auto stage = [&](int tile, int buf) {
    const int s = tile * 16 + ky;
    const float *kp, *vp;
    if (s < SINK) {                       // sink keys
      kp = kbase + (size_t)s * DH;
      vp = vbase + (size_t)s * DH;
    } else if (s < WINDOW) {              // sliding-window cache keys
      kp = kbase + (size_t)(s + 2048) * DH;
      vp = vbase + (size_t)(s + 2048) * DH;
    } else {                              // freshly projected keys
      int tt = s - WINDOW;
      if (tt > T_SZ - 1) tt = T_SZ - 1;   // clamp for masked lanes
      kp = knew + ((size_t)(b * T_SZ + tt) * NKV + kv) * DH;
      vp = vnew + ((size_t)(b * T_SZ + tt) * NKV + kv) * DH;
    }
    kp += hh * 64;
    vp += hh * 64;
    float* kd = &ktile[buf][ky * LROW + hh * 64];
    float* vd = &vtile[buf][ky * LROW + hh * 64];
#if USE_ASYNC
    #pragma unroll
    for (int d4 = 0; d4 < 16; ++d4) {
      async_cp16(kp + d4 * 4, kd + d4 * 4);
      async_cp16(vp + d4 * 4, vd + d4 * 4);
    }
#else
    #pragma unroll
    for (int d4 = 0; d4 < 16; ++d4) {
      *(float4*)(kd + d4 * 4) = *(const float4*)(kp + d4 * 4);
      *(float4*)(vd + d4 * 4) = *(const float4*)(vp + d4 * 4);
    }
#endif
  };

  v8f oacc[8];
  #pragma unroll
  for (int c = 0; c < 8; ++c) oacc[c] = (v8f){};
  float rm[8], rl[8];
  #pragma unroll
  for (int j = 0; j < 8; ++j) { rm[j] = NEG_BIG; rl[j] = 0.0f; }

  stage(0, 0);  // prologue: Q + tile0 in flight

  for (int tile = 0; tile < NTILES; ++tile) {
    const int cur = tile & 1;
    if (tile + 1 < NTILES) {
      stage(tile + 1, cur ^ 1);          // overlap next tile's DMA
#if USE_ASYNC
      __builtin_amdgcn_s_wait_asynccnt(32);   // 32 newest may remain in flight
#endif
    } else {
#if USE_ASYNC
      __builtin_amdgcn_s_wait_asynccnt(0);
#endif
    }
    __syncthreads();

    // ----- scores: S(16x16) = Qhat(16x128) @ K^T(128x16) -----
    v8f sacc = {};
    #pragma unroll
    for (int kb = 0; kb < 32; ++kb) {
      const float* ql = &qtile[l16 * LROW + kb * 4 + 2 * half];
      const float* kl = &ktile[cur][l16 * LROW + kb * 4 + 2 * half];
      v2f af, bf;
      af.x = ql[0];
      af.y = ql[1];
      bf.x = kl[0];
      bf.y = kl[1];
      sacc = __builtin_amdgcn_wmma_f32_16x16x4_f32(
          false, af, false, bf, (short)0, sacc, false, false);
    }

    // mask out-of-range key columns (col = l16)
    if (tile * 16 + l16 >= SEFF) {
      #pragma unroll
      for (int j = 0; j < 8; ++j) sacc[j] = NEG_BIG;
    }

    // ----- online softmax update (row j+8*half lives across 16 lanes) -----
    #pragma unroll
    for (int j = 0; j < 8; ++j) {
      float tm = sacc[j];
      tm = fmaxf(tm, __shfl_xor(tm, 1, 16));
      tm = fmaxf(tm, __shfl_xor(tm, 2, 16));
      tm = fmaxf(tm, __shfl_xor(tm, 4, 16));
      tm = fmaxf(tm, __shfl_xor(tm, 8, 16));
      float mn = fmaxf(rm[j], tm);
      float f  = __expf(rm[j] - mn);
      float p  = __expf(sacc[j] - mn);
      float ps = p;
      ps += __shfl_xor(ps, 1, 16);
      ps += __shfl_xor(ps, 2, 16);
      ps += __shfl_xor(ps, 4, 16);
      ps += __shfl_xor(ps, 8, 16);
      rl[j] = rl[j] * f + ps;
      rm[j] = mn;
      #pragma unroll
      for (int c = 0; c < 8; ++c) oacc[c][j] *= f;
      ptile[(j + 8 * half) * 17 + l16] = p;   // C/D layout -> LDS (row-major)
    }
    __syncthreads();

    // ----- O += P(16x16) @ V(16x128) -----
    #pragma unroll
    for (int c = 0; c < 8; ++c) {
      #pragma unroll
      for (int kk = 0; kk < 4; ++kk) {
        const float* pl = &ptile[l16 * 17 + kk * 4 + 2 * half];
        v2f af, bf;
        af.x = pl[0];
        af.y = pl[1];
        const float* vl = &vtile[cur][(kk * 4 + 2 * half) * LROW + c * 16 + l16];
        bf.x = vl[0];
        bf.y = vl[LROW];
        oacc[c] = __builtin_amdgcn_wmma_f32_16x16x4_f32(
            false, af, false, bf, (short)0, oacc[c], false, false);
      }
    }
    __syncthreads();  // buffer `cur` free for reuse at tile+2
  }

  // ----- epilogue: y[b, t, head, :] = O / l -----
  #pragma unroll
  for (int j = 0; j < 8; ++j) {
    const int M  = j + 8 * half;
    const int tt = M & 3, gg = M >> 2;
    const float inv = 1.0f / rl[j];
    float* yr = y + (size_t)(b * T_SZ + tt) * C_SZ + (kv * GROUP + gg) * DH;
    #pragma unroll
    for (int c = 0; c < 8; ++c) yr[c * 16 + l16] = oacc[c][j] * inv;
  }
}

// ---------------------------------------------------------------------------
// Launch
// ---------------------------------------------------------------------------
extern "C" void kernel_launch(void* const* d_in, const int* in_sizes, int n_in,
                              void* d_out, int out_size, void* d_ws,
                              size_t ws_size, hipStream_t stream) {
  const float* x       = (const float*)d_in[0];
  const float* cache_k = (const float*)d_in[1];
  const float* cache_v = (const float*)d_in[2];
  const float* wq      = (const float*)d_in[3];
  const float* wk      = (const float*)d_in[4];
  const float* wv      = (const float*)d_in[5];
  const float* wproj   = (const float*)d_in[6];
  const int*   spos    = (const int*)d_in[7];
  float* out = (float*)d_out;

  float* ws = (float*)d_ws;
  float* qb = ws;                       // 64 x 2048
  float* kb = ws + 131072;              // 64 x 512
  float* vb = ws + 131072 + 32768;      // 64 x 512
  float* yb = ws + 131072 + 65536;      // 64 x 2048

  dim3 wave(32);
  // QKV projections
  wmma_gemm_f32<C_SZ><<<dim3(C_SZ / 16, 4), wave, 0, stream>>>(x, wq, qb);
  wmma_gemm_f32<512><<<dim3(512 / 16, 4), wave, 0, stream>>>(x, wk, kb);
  wmma_gemm_f32<512><<<dim3(512 / 16, 4), wave, 0, stream>>>(x, wv, vb);
  // RoPE on q (folding in 1/sqrt(dh)) and k
  rope_kernel<<<(64 * NH * 64) / 256, 256, 0, stream>>>(
      qb, NH, spos, 0.08838834764831845f);
  rope_kernel<<<(64 * NKV * 64) / 256, 256, 0, stream>>>(kb, NKV, spos, 1.0f);
  // GQA flash attention over sink + sliding window
  attn_kernel<<<B_SZ * NKV, wave, 0, stream>>>(qb, kb, vb, cache_k, cache_v,
                                               yb);
  // output projection
  wmma_gemm_f32<C_SZ><<<dim3(C_SZ / 16, 4), wave, 0, stream>>>(yb, wproj, out);
}